// My_GAT_47399259079074
// MI455X (gfx1250) — compile-verified
//
#include <hip/hip_runtime.h>
#include <math.h>

typedef __attribute__((ext_vector_type(2))) float v2f;
typedef __attribute__((ext_vector_type(8))) float v8f;

#define LEAKY_SLOPE 0.01f

// ---------------------------------------------------------------------------
// WMMA fp32 GEMM:  out[N x (16*JTILES)] = A[N x KDIM] @ W[(16*JTILES) x KDIM]^T (+bias)
// One wave32 computes one 16x16 output tile using V_WMMA_F32_16X16X4_F32.
// N must be a multiple of 16 (N=50000=3125*16 holds).
// A-matrix layout (16x4 f32): lane<16 holds row m=lane, K={kk,kk+1}; lane>=16 K={kk+2,kk+3}.
// B-matrix layout (4x16 f32): mirrored; B[k][n] = W[j0+n][kk+k].
// C/D layout: c[v] = out[r0 + v + 8*(lane>>4)][j0 + (lane&15)].
// ---------------------------------------------------------------------------
template <int KDIM, int JTILES>
__global__ __launch_bounds__(256) void gemm_wmma_f32(
    const float* __restrict__ A, const float* __restrict__ W,
    const float* __restrict__ bias, float* __restrict__ out, int ntiles) {
  const int wave = (int)((blockIdx.x * blockDim.x + threadIdx.x) >> 5);
  const int lane = (int)(threadIdx.x & 31);
  const int total = ntiles * JTILES;
  if (wave >= total) return;  // wave-uniform: EXEC stays all-ones for WMMA

  const int r0 = (wave / JTILES) * 16;
  const int j0 = (wave % JTILES) * 16;
  const int m = lane & 15;
  const int half = lane >> 4;

  const float* __restrict__ arow = A + (size_t)(r0 + m) * KDIM + 2 * half;
  const float* __restrict__ wrow = W + (size_t)(j0 + m) * KDIM + 2 * half;

  v8f c = {};
#pragma unroll
  for (int kk = 0; kk < KDIM; kk += 4) {
    v2f a, b;
    a.x = arow[kk];
    a.y = arow[kk + 1];
    b.x = wrow[kk];
    b.y = wrow[kk + 1];
    c = __builtin_amdgcn_wmma_f32_16x16x4_f32(false, a, false, b, (short)0, c,
                                              false, false);
  }

  const float bj = bias ? bias[j0 + m] : 0.0f;
  const int OUTD = JTILES * 16;
#pragma unroll
  for (int v = 0; v < 8; ++v) {
    out[(size_t)(r0 + v + 8 * half) * OUTD + j0 + m] = c[v] + bj;
  }
}

// ---------------------------------------------------------------------------
// CSR row_start from sorted dst (every node occurs at least once).
// ---------------------------------------------------------------------------
__global__ void build_rowstart(const int* __restrict__ dst,
                               int* __restrict__ row_start, int ecount, int n) {
  int i = (int)(blockIdx.x * blockDim.x + threadIdx.x);
  if (i == 0) row_start[n] = ecount;
  if (i >= ecount) return;
  if (i == 0 || dst[i] != dst[i - 1]) row_start[dst[i]] = i;
}

// ---------------------------------------------------------------------------
// Per-node attention projections: s_src[n] = z[n]·a_src, s_dst[n] = z[n]·a_dst
// ---------------------------------------------------------------------------
__global__ void node_scores(const float* __restrict__ z,
                            const float* __restrict__ Wa,
                            float* __restrict__ s_src,
                            float* __restrict__ s_dst, int n) {
  int i = (int)(blockIdx.x * blockDim.x + threadIdx.x);
  if (i >= n) return;
  const float* __restrict__ zr = z + (size_t)i * 64;
  float ss = 0.f, sd = 0.f;
#pragma unroll 8
  for (int f = 0; f < 64; ++f) {
    float zv = zr[f];
    ss += zv * Wa[f];
    sd += zv * Wa[64 + f];
  }
  s_src[i] = ss;
  s_dst[i] = sd;
}

// w @ a_w is affine in e_w:  e_w[i]*c0 + c1  (rank-1 edge embedding)
__global__ void edge_const(const float* __restrict__ W_emb_e,
                           const float* __restrict__ b_emb_e,
                           const float* __restrict__ Wa,
                           float* __restrict__ c01) {
  if (threadIdx.x == 0 && blockIdx.x == 0) {
    float c0 = 0.f, c1 = 0.f;
    for (int f = 0; f < 64; ++f) {
      float aw = Wa[128 + f];
      c0 += W_emb_e[f] * aw;
      c1 += b_emb_e[f] * aw;
    }
    c01[0] = c0;
    c01[1] = c1;
  }
}

__global__ void edge_scores(const float* __restrict__ s_src,
                            const float* __restrict__ s_dst,
                            const int* __restrict__ src,
                            const int* __restrict__ dst,
                            const float* __restrict__ e_w,
                            const float* __restrict__ c01,
                            float* __restrict__ e, int ecount) {
  int i = (int)(blockIdx.x * blockDim.x + threadIdx.x);
  if (i >= ecount) return;
  float v = s_src[src[i]] + s_dst[dst[i]] + e_w[i] * c01[0] + c01[1];
  e[i] = v > 0.f ? v : LEAKY_SLOPE * v;
}

// ---------------------------------------------------------------------------
// One wave32 per node: segment max -> softmax denom -> weighted gather of
// z[src], then residual + relu update. Atomic-free thanks to sorted dst.
// Lanes split edges for reductions; lanes split features (2 each) for gather.
// ---------------------------------------------------------------------------
__global__ __launch_bounds__(256) void gat_aggregate(
    const float* __restrict__ hin, const float* __restrict__ hs,
    const float* __restrict__ z, const float* __restrict__ e,
    const int* __restrict__ src, const int* __restrict__ row_start,
    float* __restrict__ hout, int n) {
  const int wid = (int)((blockIdx.x * blockDim.x + threadIdx.x) >> 5);
  const int lane = (int)(threadIdx.x & 31);
  if (wid >= n) return;

  const int start = row_start[wid];
  const int end = row_start[wid + 1];

  // segment max over this node's incoming edges
  float mx = -3.402823466e38f;
  for (int j = start + lane; j < end; j += 32) mx = fmaxf(mx, e[j]);
#pragma unroll
  for (int off = 16; off; off >>= 1) mx = fmaxf(mx, __shfl_xor(mx, off, 32));

  // softmax denominator
  float sum = 0.f;
  for (int j = start + lane; j < end; j += 32) sum += expf(e[j] - mx);
#pragma unroll
  for (int off = 16; off; off >>= 1) sum += __shfl_xor(sum, off, 32);
  const float inv = 1.0f / sum;

  // weighted feature gather: lane owns features {2*lane, 2*lane+1}
  const int f0 = lane * 2;
  float a0 = 0.f, a1 = 0.f;
  for (int j = start; j < end; ++j) {
    const float coef = expf(e[j] - mx) * inv;
    const float* __restrict__ zr = z + (size_t)src[j] * 64 + f0;
    a0 += coef * zr[0];
    a1 += coef * zr[1];
  }

  const size_t base = (size_t)wid * 64 + f0;
  float t0 = hs[base] + a0;
  float t1 = hs[base + 1] + a1;
  t0 = t0 > 0.f ? t0 : 0.f;
  t1 = t1 > 0.f ? t1 : 0.f;
  hout[base] = hin[base] + t0;
  hout[base + 1] = hin[base + 1] + t1;
}

// ---------------------------------------------------------------------------
// BatchNorm statistics + fold into output GEMM weights
// ---------------------------------------------------------------------------
__global__ void zero_f(float* __restrict__ p, int n) {
  int i = (int)(blockIdx.x * blockDim.x + threadIdx.x);
  if (i < n) p[i] = 0.f;
}

__global__ __launch_bounds__(256) void bn_stats(const float* __restrict__ h,
                                                float* __restrict__ stats,
                                                int n) {
  const int f = (int)(threadIdx.x & 63);
  const int rl = (int)(threadIdx.x >> 6);
  float s = 0.f, q = 0.f;
  for (int r = (int)blockIdx.x * 4 + rl; r < n; r += (int)gridDim.x * 4) {
    float v = h[(size_t)r * 64 + f];
    s += v;
    q += v * v;
  }
  atomicAdd(&stats[f], s);
  atomicAdd(&stats[64 + f], q);
}

__global__ void bn_finalize(const float* __restrict__ stats,
                            const float* __restrict__ gamma,
                            const float* __restrict__ beta,
                            float* __restrict__ sf, float* __restrict__ tf,
                            float ninv) {
  int f = (int)threadIdx.x;
  if (f < 64) {
    float mu = stats[f] * ninv;
    float var = stats[64 + f] * ninv - mu * mu;
    float sc = gamma[f] * rsqrtf(var + 1e-5f);
    sf[f] = sc;
    tf[f] = beta[f] - mu * sc;
  }
}

__global__ void make_out_weights(const float* __restrict__ W_out,
                                 const float* __restrict__ b_out,
                                 const float* __restrict__ sf,
                                 const float* __restrict__ tf,
                                 float* __restrict__ Wmod,
                                 float* __restrict__ bmod) {
  int i = (int)(blockIdx.x * blockDim.x + threadIdx.x);
  if (i < 32 * 64) {
    int f = i & 63;
    Wmod[i] = W_out[i] * sf[f];
  }
  if (i < 32) {
    float b = b_out[i];
    for (int f = 0; f < 64; ++f) b += tf[f] * W_out[i * 64 + f];
    bmod[i] = b;
  }
}

// ---------------------------------------------------------------------------
extern "C" void kernel_launch(void* const* d_in, const int* in_sizes, int n_in,
                              void* d_out, int out_size, void* d_ws,
                              size_t ws_size, hipStream_t stream) {
  const float* h_in = (const float*)d_in[0];     // N x 32
  const float* e_w = (const float*)d_in[1];      // E x 1
  const int* src = (const int*)d_in[4];          // E
  const int* dst = (const int*)d_in[5];          // E (sorted)
  const float* W_emb_h = (const float*)d_in[6];  // 64 x 32
  const float* b_emb_h = (const float*)d_in[7];  // 64
  const float* W_emb_e = (const float*)d_in[8];  // 64 x 1
  const float* b_emb_e = (const float*)d_in[9];  // 64
  const float* Ws1 = (const float*)d_in[10];
  const float* Wf1 = (const float*)d_in[11];
  const float* Wa1 = (const float*)d_in[12];  // 1 x 192
  const float* Ws2 = (const float*)d_in[13];
  const float* Wf2 = (const float*)d_in[14];
  const float* Wa2 = (const float*)d_in[15];
  const float* gamma = (const float*)d_in[16];
  const float* beta = (const float*)d_in[17];
  const float* W_out = (const float*)d_in[18];  // 32 x 64
  const float* b_out = (const float*)d_in[19];  // 32

  const int N = in_sizes[0] / 32;  // 50000
  const int E = in_sizes[4];       // 800000
  const int ntiles = N / 16;       // 3125

  // workspace layout (floats)
  float* ws = (float*)d_ws;
  float* bufA = ws;                    // N*64
  float* bufB = bufA + (size_t)N * 64; // N*64
  float* zbuf = bufB + (size_t)N * 64; // N*64
  float* ebuf = zbuf + (size_t)N * 64; // E
  float* ssrc = ebuf + E;              // N
  float* sdst = ssrc + N;              // N
  int* rowstart = (int*)(sdst + N);    // N+1 (pad to N+2)
  float* c01 = (float*)(rowstart + N + 2);  // 2
  float* stats = c01 + 2;                   // 128 (sum | sumsq)
  float* sf = stats + 128;                  // 64
  float* tf = sf + 64;                      // 64
  float* Wmod = tf + 64;                    // 32*64
  float* bmod = Wmod + 32 * 64;             // 32

  const int gemm64_blocks = (ntiles * 4 + 7) / 8;   // waves: ntiles*4, 8/block
  const int gemm32o_blocks = (ntiles * 2 + 7) / 8;  // output gemm
  const int nodeBlocks = (N + 255) / 256;
  const int edgeBlocks = (E + 255) / 256;
  const int aggBlocks = (N + 7) / 8;  // one wave per node

  // CSR build (dst fixed across both layers)
  build_rowstart<<<edgeBlocks, 256, 0, stream>>>(dst, rowstart, E, N);

  // Embedding: h0 = h @ W_emb_h^T + b_emb_h   -> bufA
  gemm_wmma_f32<32, 4><<<gemm64_blocks, 256, 0, stream>>>(
      h_in, W_emb_h, b_emb_h, bufA, ntiles);

  // ----- GAT layer 1: bufA -> bufB -----
  gemm_wmma_f32<64, 4><<<gemm64_blocks, 256, 0, stream>>>(bufA, Ws1, nullptr,
                                                          bufB, ntiles);
  gemm_wmma_f32<64, 4><<<gemm64_blocks, 256, 0, stream>>>(bufA, Wf1, nullptr,
                                                          zbuf, ntiles);
  node_scores<<<nodeBlocks, 256, 0, stream>>>(zbuf, Wa1, ssrc, sdst, N);
  edge_const<<<1, 64, 0, stream>>>(W_emb_e, b_emb_e, Wa1, c01);
  edge_scores<<<edgeBlocks, 256, 0, stream>>>(ssrc, sdst, src, dst, e_w, c01,
                                              ebuf, E);
  gat_aggregate<<<aggBlocks, 256, 0, stream>>>(bufA, bufB, zbuf, ebuf, src,
                                               rowstart, bufB, N);

  // ----- GAT layer 2: bufB -> bufA -----
  gemm_wmma_f32<64, 4><<<gemm64_blocks, 256, 0, stream>>>(bufB, Ws2, nullptr,
                                                          bufA, ntiles);
  gemm_wmma_f32<64, 4><<<gemm64_blocks, 256, 0, stream>>>(bufB, Wf2, nullptr,
                                                          zbuf, ntiles);
  node_scores<<<nodeBlocks, 256, 0, stream>>>(zbuf, Wa2, ssrc, sdst, N);
  edge_const<<<1, 64, 0, stream>>>(W_emb_e, b_emb_e, Wa2, c01);
  edge_scores<<<edgeBlocks, 256, 0, stream>>>(ssrc, sdst, src, dst, e_w, c01,
                                              ebuf, E);
  gat_aggregate<<<aggBlocks, 256, 0, stream>>>(bufB, bufA, zbuf, ebuf, src,
                                               rowstart, bufA, N);

  // ----- BatchNorm folded into output GEMM -----
  zero_f<<<1, 128, 0, stream>>>(stats, 128);
  bn_stats<<<512, 256, 0, stream>>>(bufA, stats, N);
  bn_finalize<<<1, 64, 0, stream>>>(stats, gamma, beta, sf, tf, 1.0f / N);
  make_out_weights<<<8, 256, 0, stream>>>(W_out, b_out, sf, tf, Wmod, bmod);

  // out = normalized(h) @ W_out^T + b_out == bufA @ Wmod^T + bmod
  gemm_wmma_f32<64, 2><<<gemm32o_blocks, 256, 0, stream>>>(
      bufA, Wmod, bmod, (float*)d_out, ntiles);
}